// opencvemd_26336739459366
// MI455X (gfx1250) — compile-verified
//
#include <hip/hip_runtime.h>
#include <math.h>

// Sinkhorn OT, B=32, M=2048, D=2, 50 iters, EPS=0.02.
// C is never materialized: rank-2 cross term via V_WMMA_F32_16X16X4_F32,
// diagonal terms + dual potentials folded into per-row/per-col scalars.
// f/g ping-pong through d_ws; 100 update launches for occupancy.
// Final argmax pass split 8-ways per batch with deterministic partial sums.

#define BATCH 32
#define MPTS  2048
#define ITERS 50
#define FPARTS 8   // final-pass blocks per batch

// KS = log2(e)/EPS ; exponent computed in log2 units so exp2f/log2f map to
// v_exp_f32 / v_log_f32 directly.
#define KS       72.13475204444817f
#define TWO_KS   144.26950408889634f
#define EPS_LOGW (-0.15249237961f)   /* EPS * (-ln 2048) */
#define EPS_LN2  (0.01386294361f)    /* EPS * ln 2       */

typedef __attribute__((ext_vector_type(2))) float v2f;
typedef __attribute__((ext_vector_type(8))) float v8f;

__device__ __forceinline__ float groupMax16(float v) {
#pragma unroll
  for (int m = 1; m < 16; m <<= 1) v = fmaxf(v, __shfl_xor(v, m, 32));
  return v;
}
__device__ __forceinline__ float groupSum16(float v) {
#pragma unroll
  for (int m = 1; m < 16; m <<= 1) v += __shfl_xor(v, m, 32);
  return v;
}

__global__ void sink_init(float* __restrict__ fg) {
  int i = blockIdx.x * blockDim.x + threadIdx.x;
  if (i < 2 * BATCH * MPTS) fg[i] = 0.0f;
}

// One half-iteration: dualP[i] = EPS*log_w - EPS*logsumexp_j((dualQ[j]-C[i,j])/EPS)
// P = "row side" points [B,M,2], Q = "column side" points [B,M,2].
// grid: (16 row-groups, 32 batches), block: 256 threads = 8 waves.
// Each wave owns a 16-row WMMA block; 128 column tiles of 16.
__global__ __launch_bounds__(256) void sink_update(
    const float* __restrict__ P, const float* __restrict__ Q,
    const float* __restrict__ dualQ, float* __restrict__ dualP) {
  __shared__ float s_qx[MPTS], s_qy[MPTS], s_hs[MPTS];

  const int b   = blockIdx.y;
  const int bo  = b * MPTS;
  const int tid = threadIdx.x;

  // Stage column-side points and combined column term into LDS.
  for (int j = tid; j < MPTS; j += 256) {
    float qx = Q[(bo + j) * 2 + 0];
    float qy = Q[(bo + j) * 2 + 1];
    s_qx[j] = qx;
    s_qy[j] = qy;
    s_hs[j] = KS * (dualQ[bo + j] - (qx * qx + qy * qy));
  }
  __syncthreads();

  const int lane    = tid & 31;
  const int wave    = tid >> 5;
  const int rowBase = blockIdx.x * 128 + wave * 16;
  const int half    = lane >> 4;   // 0: rows M=0..7, 1: rows M=8..15 of tile
  const int l16     = lane & 15;

  // A operand (16x4 fp32): lane<16 holds K=0,1 = scaled coords; lane>=16 pads K=2,3.
  const int ra = rowBase + l16;
  const float pax = P[(bo + ra) * 2 + 0];
  const float pay = P[(bo + ra) * 2 + 1];
  v2f aReg;
  aReg.x = (lane < 16) ? TWO_KS * pax : 0.0f;
  aReg.y = (lane < 16) ? TWO_KS * pay : 0.0f;

  // Per-row term -KS*|p_i|^2 for the 8 rows this lane's D registers cover.
  float rs[8];
#pragma unroll
  for (int v = 0; v < 8; ++v) {
    int r = rowBase + v + 8 * half;
    float x = P[(bo + r) * 2 + 0];
    float y = P[(bo + r) * 2 + 1];
    rs[v] = -KS * (x * x + y * y);
  }

  // ---- pass 1: streaming row max over 128 column tiles ----
  float rm[8];
#pragma unroll
  for (int v = 0; v < 8; ++v) rm[v] = -3.4e38f;

  for (int t = 0; t < 128; ++t) {
    const int c = (t << 4) + l16;
    const float qx = s_qx[c], qy = s_qy[c], hs = s_hs[c];
    v2f bReg;
    bReg.x = (lane < 16) ? qx : 0.0f;
    bReg.y = (lane < 16) ? qy : 0.0f;
    v8f cz = {};
    v8f d = __builtin_amdgcn_wmma_f32_16x16x4_f32(
        false, aReg, false, bReg, (short)0, cz, false, false);
#pragma unroll
    for (int v = 0; v < 8; ++v) rm[v] = fmaxf(rm[v], d[v] + hs + rs[v]);
  }
#pragma unroll
  for (int v = 0; v < 8; ++v) rm[v] = groupMax16(rm[v]);

  // ---- pass 2: exp2-sum with the stabilizing max subtracted ----
  float ssum[8];
#pragma unroll
  for (int v = 0; v < 8; ++v) ssum[v] = 0.0f;

  for (int t = 0; t < 128; ++t) {
    const int c = (t << 4) + l16;
    const float qx = s_qx[c], qy = s_qy[c], hs = s_hs[c];
    v2f bReg;
    bReg.x = (lane < 16) ? qx : 0.0f;
    bReg.y = (lane < 16) ? qy : 0.0f;
    v8f cz = {};
    v8f d = __builtin_amdgcn_wmma_f32_16x16x4_f32(
        false, aReg, false, bReg, (short)0, cz, false, false);
#pragma unroll
    for (int v = 0; v < 8; ++v) ssum[v] += exp2f(d[v] + hs + rs[v] - rm[v]);
  }
#pragma unroll
  for (int v = 0; v < 8; ++v) ssum[v] = groupSum16(ssum[v]);

  if (l16 == 0) {
#pragma unroll
    for (int v = 0; v < 8; ++v) {
      int r = rowBase + v + 8 * half;
      dualP[bo + r] = EPS_LOGW - EPS_LN2 * (rm[v] + log2f(ssum[v]));
    }
  }
}

// Final pass, part 1: per row i, j* = argmax_j (g[j] - C[i,j]) (f[i] is
// row-constant so it never affects the argmax), accumulate geometric
// ||p1_i - p2_{j*}||^2. grid (FPARTS, BATCH), 1 row per thread, deterministic
// block reduction into a per-(batch,part) partial sum.
__global__ __launch_bounds__(256) void sink_final_part(
    const float* __restrict__ P1, const float* __restrict__ P2,
    const float* __restrict__ g, float* __restrict__ partial) {
  __shared__ float s_qx[MPTS], s_qy[MPTS], s_h[MPTS];
  __shared__ float s_red[256];

  const int b   = blockIdx.y;
  const int bo  = b * MPTS;
  const int tid = threadIdx.x;

  for (int j = tid; j < MPTS; j += 256) {
    float qx = P2[(bo + j) * 2 + 0];
    float qy = P2[(bo + j) * 2 + 1];
    s_qx[j] = qx;
    s_qy[j] = qy;
    s_h[j]  = g[bo + j] - (qx * qx + qy * qy);
  }
  __syncthreads();

  const int r = blockIdx.x * 256 + tid;   // one row per thread
  const float px = P1[(bo + r) * 2 + 0];
  const float py = P1[(bo + r) * 2 + 1];
  float best = -3.4e38f;
  int bj = 0;
  for (int j = 0; j < MPTS; ++j) {
    float v = s_h[j] + 2.0f * (px * s_qx[j] + py * s_qy[j]);
    if (v > best) { best = v; bj = j; }  // strict > => first max (jnp.argmax)
  }
  const float dx = px - s_qx[bj];
  const float dy = py - s_qy[bj];

  s_red[tid] = dx * dx + dy * dy;
  __syncthreads();
  for (int s = 128; s > 0; s >>= 1) {
    if (tid < s) s_red[tid] += s_red[tid + s];
    __syncthreads();
  }
  if (tid == 0) partial[b * FPARTS + blockIdx.x] = s_red[0];
}

// Final pass, part 2: fold the FPARTS partial sums per batch (deterministic).
__global__ void sink_final_reduce(const float* __restrict__ partial,
                                  float* __restrict__ out) {
  int b = blockIdx.x * blockDim.x + threadIdx.x;
  if (b < BATCH) {
    float s = 0.0f;
#pragma unroll
    for (int p = 0; p < FPARTS; ++p) s += partial[b * FPARTS + p];
    out[b] = s;
  }
}

extern "C" void kernel_launch(void* const* d_in, const int* in_sizes, int n_in,
                              void* d_out, int out_size, void* d_ws, size_t ws_size,
                              hipStream_t stream) {
  (void)in_sizes; (void)n_in; (void)out_size; (void)ws_size;
  const float* p1 = (const float*)d_in[0];
  const float* p2 = (const float*)d_in[1];
  float* out  = (float*)d_out;
  float* fbuf = (float*)d_ws;               // [B*M] f potential
  float* gbuf = fbuf + BATCH * MPTS;        // [B*M] g potential
  float* part = gbuf + BATCH * MPTS;        // [B*FPARTS] final partial sums

  sink_init<<<(2 * BATCH * MPTS + 255) / 256, 256, 0, stream>>>(fbuf);

  dim3 ugrid(16, BATCH);
  for (int it = 0; it < ITERS; ++it) {
    sink_update<<<ugrid, 256, 0, stream>>>(p1, p2, gbuf, fbuf);  // f-update
    sink_update<<<ugrid, 256, 0, stream>>>(p2, p1, fbuf, gbuf);  // g-update
  }

  dim3 fgrid(FPARTS, BATCH);
  sink_final_part<<<fgrid, 256, 0, stream>>>(p1, p2, gbuf, part);
  sink_final_reduce<<<1, BATCH, 0, stream>>>(part, out);
}